// SNN_25125558682114
// MI455X (gfx1250) — compile-verified
//
#include <hip/hip_runtime.h>
#include <hip/hip_bf16.h>

typedef __attribute__((ext_vector_type(2))) float v2f;
typedef __attribute__((ext_vector_type(8))) float v8f;

#define T_STEPS 128
#define BATCH   64
#define NI      784
#define NH      512
#define NO      10

// ---------------------------------------------------------------------------
// GEMM1: I_h[m][h] = sum_n data[b][t][n] * W_h[h][n],  m = t*BATCH + b
// One wave per 16x16 output tile, K-loop of 196 steps of V_WMMA_F32_16X16X4_F32.
// ---------------------------------------------------------------------------
__global__ __launch_bounds__(128) void snn_gemm_hidden(
    const float* __restrict__ data,   // [B, T, NI]
    const float* __restrict__ Wh,     // [NH, NI]
    float* __restrict__ Ih)           // [T, B, NH] == [8192, NH]
{
    const int tile  = blockIdx.x * 4 + (threadIdx.x >> 5);   // 16384 tiles
    const int nTile = tile & 31;                              // NH/16 = 32
    const int mTile = tile >> 5;                              // 8192/16 = 512
    const int m0 = mTile << 4;
    const int h0 = nTile << 4;

    const int lane = threadIdx.x & 31;
    const int r    = lane & 15;
    const int half = lane >> 4;

    const int m = m0 + r;                     // output row for this lane
    const int t = m >> 6;                     // m / BATCH
    const int b = m & 63;                     // m % BATCH
    const float* __restrict__ arow = data + ((size_t)(b * T_STEPS + t)) * NI + 2 * half;
    const float* __restrict__ brow = Wh   + ((size_t)(h0 + r)) * NI + 2 * half;

    v8f acc = {};
    for (int k0 = 0; k0 < NI; k0 += 4) {
        v2f a = *(const v2f*)(arow + k0);
        v2f w = *(const v2f*)(brow + k0);
        acc = __builtin_amdgcn_wmma_f32_16x16x4_f32(
            false, a, false, w, (short)0, acc, false, false);
    }

    float* __restrict__ outBase = Ih + (size_t)(m0 + 8 * half) * NH + h0 + r;
    #pragma unroll
    for (int v = 0; v < 8; ++v)
        outBase[(size_t)v * NH] = acc[v];
}

// ---------------------------------------------------------------------------
// GEMM2: I_o[m][o] = sum_h spk_h[m][h] * W_o[o][h]  (N padded 10 -> 16)
// ---------------------------------------------------------------------------
__global__ __launch_bounds__(128) void snn_gemm_out(
    const float* __restrict__ spk,    // [8192, NH]
    const float* __restrict__ Wo,     // [NO, NH]
    float* __restrict__ Io)           // [8192, NO]
{
    const int tile = blockIdx.x * 4 + (threadIdx.x >> 5);    // 512 tiles
    const int m0   = tile << 4;

    const int lane = threadIdx.x & 31;
    const int r    = lane & 15;
    const int half = lane >> 4;

    const float colMask = (r < NO) ? 1.0f : 0.0f;   // kill padded columns, no EXEC change
    const int   rc      = (r < NO) ? r : (NO - 1);  // clamp pointer, avoid OOB

    const float* __restrict__ arow = spk + ((size_t)(m0 + r)) * NH + 2 * half;
    const float* __restrict__ brow = Wo  + ((size_t)rc) * NH + 2 * half;

    v8f acc = {};
    for (int k0 = 0; k0 < NH; k0 += 4) {
        v2f a = *(const v2f*)(arow + k0);
        v2f w = *(const v2f*)(brow + k0);
        w.x *= colMask;
        w.y *= colMask;
        acc = __builtin_amdgcn_wmma_f32_16x16x4_f32(
            false, a, false, w, (short)0, acc, false, false);
    }

    if (r < NO) {
        float* __restrict__ outBase = Io + (size_t)(m0 + 8 * half) * NO + r;
        #pragma unroll
        for (int v = 0; v < 8; ++v)
            outBase[(size_t)v * NO] = acc[v];
    }
}

// ---------------------------------------------------------------------------
// Fractional LIF scan.  One thread per neuron, P = B*H neurons total.
// History kept in LDS as hist[k][tid] (stride-1 across lanes -> no bank
// conflicts).  No barriers in the time loop: each thread owns its column.
//   V_n = KAPPA*(-G_L*(V-V_L)+I)/CM + V - sum_{j>=1} c_j * dV_{n-1-j}
// ---------------------------------------------------------------------------
#define FLIF_THREADS 64

__global__ __launch_bounds__(FLIF_THREADS) void snn_flif(
    const float* __restrict__ I,      // [T, P]
    float* __restrict__ spkOut,       // [T, P]
    float* __restrict__ memOut,       // [T, P] or nullptr
    int P)
{
    __shared__ float cj[T_STEPS + 1];
    __shared__ float hist[T_STEPS * FLIF_THREADS];   // 32 KB

    const int tid = threadIdx.x;
    const int p   = blockIdx.x * FLIF_THREADS + tid;

    // c_j = (j+1)^{1-a} - j^{1-a},  1-a = 0.8
    for (int j = tid; j <= T_STEPS; j += FLIF_THREADS)
        cj[j] = __powf((float)(j + 1), 0.8f) - __powf((float)j, 0.8f);
    __syncthreads();

    const float KAPPA = 0.58766347f;   // 0.1^0.2 * Gamma(1.8)
    float V = -70.0f;                  // V_RESET

    for (int n = 1; n <= T_STEPS; ++n) {
        // memory term: sum over history slots k = 0 .. n-2, lag j = n-1-k
        float memory = 0.0f;
        for (int k = 0; k <= n - 2; ++k)
            memory += cj[n - 1 - k] * hist[k * FLIF_THREADS + tid];

        const float In = I[(size_t)(n - 1) * P + p];
        const float f  = (-0.025f * (V + 70.0f) + In) * 2.0f;   // /CM, CM=0.5
        const float Vn = KAPPA * f + V - memory;
        const float s  = (Vn >= -50.0f) ? 1.0f : 0.0f;
        const float Vr = (s > 0.0f) ? -70.0f : Vn;

        hist[(n - 1) * FLIF_THREADS + tid] = Vr - V;

        spkOut[(size_t)(n - 1) * P + p] = s;
        if (memOut) memOut[(size_t)(n - 1) * P + p] = Vr;
        V = Vr;
    }
}

// ---------------------------------------------------------------------------
extern "C" void kernel_launch(void* const* d_in, const int* in_sizes, int n_in,
                              void* d_out, int out_size, void* d_ws, size_t ws_size,
                              hipStream_t stream) {
    const float* data = (const float*)d_in[0];   // [B, T, NI]
    const float* Wh   = (const float*)d_in[1];   // [NH, NI]
    const float* Wo   = (const float*)d_in[2];   // [NO, NH]
    float* out = (float*)d_out;                  // spk_o [T,B,NO] ++ mem_o [T,B,NO]

    const size_t M  = (size_t)T_STEPS * BATCH;   // 8192
    float* Ih   = (float*)d_ws;                  // [M, NH]  16 MB
    float* spkH = Ih + M * NH;                   // [M, NH]  16 MB
    float* Io   = spkH + M * NH;                 // [M, NO]  0.32 MB

    // 1) hidden synapses: WMMA GEMM, 16384 tiles, 4 waves/block
    snn_gemm_hidden<<<4096, 128, 0, stream>>>(data, Wh, Ih);

    // 2) hidden fractional-LIF scan: 32768 neurons
    snn_flif<<<(BATCH * NH) / FLIF_THREADS, FLIF_THREADS, 0, stream>>>(
        Ih, spkH, nullptr, BATCH * NH);

    // 3) output synapses: WMMA GEMM (N padded to 16), 512 tiles
    snn_gemm_out<<<128, 128, 0, stream>>>(spkH, Wo, Io);

    // 4) output fractional-LIF scan: 640 neurons; write both traces to d_out
    snn_flif<<<(BATCH * NO) / FLIF_THREADS, FLIF_THREADS, 0, stream>>>(
        Io, out, out + (size_t)T_STEPS * BATCH * NO, BATCH * NO);
}